// ProCoLoss_37340445671958
// MI455X (gfx1250) — compile-verified
//
#include <hip/hip_runtime.h>
#include <math.h>

typedef __attribute__((ext_vector_type(2))) float v2f;
typedef __attribute__((ext_vector_type(8))) float v8f;

#define FDIM 256
#define NCLS 100
#define NCLS_PAD 112          // 7 tiles of 16
#define NU 127                // FDIM/2 - 1
#define INV_T 10.0f           // 1/TEMPERATURE

// ---------- device helpers ----------------------------------------------

// log(I0(x) * e^{-x}) via Abramowitz & Stegun 9.8.1 / 9.8.2
__device__ __forceinline__ float log_i0e(float x) {
  if (x < 3.75f) {
    float t = (x * x) / (3.75f * 3.75f);
    float I0 = 1.0f + t * (3.5156229f + t * (3.0899424f + t * (1.2067492f +
               t * (0.2659732f + t * (0.0360768f + t * 0.0045813f)))));
    return logf(I0) - x;
  }
  float t = 3.75f / x;
  float p = 0.39894228f + t * (0.01328592f + t * (0.00225319f + t * (-0.00157565f +
            t * (0.00916281f + t * (-0.02057706f + t * (0.02635537f +
            t * (-0.01647633f + t * 0.00392377f)))))));
  return logf(p) - 0.5f * logf(x);
}

// log(I_nu(x) e^{-x}) via Miller downward recurrence carried in log-space
// (identical math to the reference). lt[i] = log(2*i), i in [1, 2*NU].
__device__ __forceinline__ float log_ive_nu(float x, const float* lt) {
  float lx = logf(x + 1e-30f);
  float L = 0.0f;          // L_{i}
  float Lp = -INFINITY;    // L_{i+1}
  float Lnu = 0.0f;
  #pragma unroll 2
  for (int i = 2 * NU; i >= 1; --i) {
    float t = lt[i] - lx + L;              // log(2i/x) + L_i
    float m = fmaxf(t, Lp);
    float d = fminf(t, Lp) - m;            // <= 0
    float Ln = m + log1pf(__expf(d));      // L_{i-1} = logaddexp(t, Lp)
    if (i == NU + 1) Lnu = Ln;             // L_{NU}
    Lp = L;
    L = Ln;                                // after i==1: L == L_0
  }
  return log_i0e(x) + Lnu - L;
}

// ---------- kernel 1: per-class statistics -------------------------------
// One block per (padded) class; deterministic fixed-order scan over samples.
__global__ __launch_bounds__(256) void class_stats_kernel(
    const float* __restrict__ F, const float* __restrict__ Ave,
    const float* __restrict__ Amount, const int* __restrict__ labels,
    float* __restrict__ mu, float* __restrict__ kappa, int N) {
  int c = blockIdx.x, tid = threadIdx.x;
  if (c >= NCLS) {                         // pad rows for 16-wide WMMA tiles
    mu[c * FDIM + tid] = 0.0f;
    if (tid == 0) kappa[c] = 0.0f;
    return;
  }
  float s = 0.0f;
  int cnt = 0;
  for (int n = 0; n < N; ++n) {            // block-uniform branch
    if (labels[n] == c) { s += F[n * FDIM + tid]; ++cnt; }
  }
  float fc = (float)cnt;
  float ave = s / fmaxf(fc, 1.0f);
  float denom = fc + Amount[c];
  float w = (denom > 0.0f) ? (fc / denom) : 0.0f;
  float an = Ave[c * FDIM + tid] * (1.0f - w) + ave * w;

  __shared__ float red[256];
  red[tid] = an * an;
  __syncthreads();
  for (int st = 128; st > 0; st >>= 1) {
    if (tid < st) red[tid] += red[tid + st];
    __syncthreads();
  }
  float R2 = red[0];
  float R = sqrtf(R2);
  mu[c * FDIM + tid] = an / fmaxf(R, 1e-12f);
  if (tid == 0) {
    float k = (float)FDIM * R / (1.0f - R2);
    if (k > 100000.0f || k < 0.0f) k = 100000.0f;
    kappa[c] = k;
  }
}

// ---------- kernel 2: logc per class -------------------------------------
__global__ __launch_bounds__(128) void logc_kernel(
    const float* __restrict__ kappa, float* __restrict__ logc) {
  __shared__ float lt[2 * NU + 1];
  for (int i = threadIdx.x; i <= 2 * NU; i += 128) lt[i] = logf(2.0f * (float)i);
  __syncthreads();
  int c = threadIdx.x;
  if (c >= NCLS_PAD) return;
  if (c >= NCLS) { logc[c] = 0.0f; return; }
  float x = kappa[c];
  float lr0 = log_ive_nu(x, lt);
  logc[c] = lr0 + x - (float)NU * logf(x + 1e-20f);
}

// ---------- kernel 3: dot = features @ mu^T via f32 WMMA -----------------
// One wave per 16x16 output tile; K-loop of 64 x V_WMMA_F32_16X16X4_F32.
__global__ __launch_bounds__(32) void gemm_dot_kernel(
    const float* __restrict__ F, const float* __restrict__ mu,
    float* __restrict__ dot) {
  int rowTile = blockIdx.x;                // N/16 tiles of samples
  int colTile = blockIdx.y;                // 7 tiles of classes
  int lane = threadIdx.x;
  int lo = lane & 15, hi = lane >> 4;

  // A 16x4 f32: lane(lo)=M row; lanes 16-31 hold K=2,3; regs hold K,K+1
  const float* pa = F  + (rowTile * 16 + lo) * FDIM + 2 * hi;
  // B 4x16 f32: lane(lo)=N col; B[k][n] = mu[n][k]
  const float* pb = mu + (colTile * 16 + lo) * FDIM + 2 * hi;

  v8f acc = {};
  #pragma unroll 8
  for (int k0 = 0; k0 < FDIM; k0 += 4) {
    v2f a = *(const v2f*)(pa + k0);
    v2f b = *(const v2f*)(pb + k0);
    acc = __builtin_amdgcn_wmma_f32_16x16x4_f32(
        /*neg_a=*/false, a, /*neg_b=*/false, b,
        /*c_mod=*/(short)0, acc, /*reuse_a=*/false, /*reuse_b=*/false);
  }
  // D 16x16 f32: VGPR j -> row j + 8*hi, col = lo
  float* outp = dot + (rowTile * 16) * NCLS_PAD + colTile * 16;
  #pragma unroll
  for (int j = 0; j < 8; ++j)
    outp[(j + 8 * hi) * NCLS_PAD + lo] = acc[j];
}

// ---------- kernel 4: per-sample logits + fused log-softmax NLL ----------
__global__ __launch_bounds__(128) void sample_kernel(
    const float* __restrict__ F, const float* __restrict__ dot,
    const float* __restrict__ kappa, const float* __restrict__ logc,
    const int* __restrict__ labels, float* __restrict__ loss) {
  int n = blockIdx.x, tid = threadIdx.x;
  __shared__ float lt[2 * NU + 1];
  __shared__ float lg[128];
  __shared__ float red[128];

  // ||f_n||^2 (reference normalizes, but compute it faithfully)
  float v0 = F[n * FDIM + tid];
  float v1 = F[n * FDIM + 128 + tid];
  red[tid] = v0 * v0 + v1 * v1;
  for (int i = tid; i <= 2 * NU; i += 128) lt[i] = logf(2.0f * (float)i);
  __syncthreads();
  for (int st = 64; st > 0; st >>= 1) {
    if (tid < st) red[tid] += red[tid + st];
    __syncthreads();
  }
  float fn2 = red[0];
  __syncthreads();

  float l = -INFINITY;
  if (tid < NCLS) {
    float k = kappa[tid];
    float d = dot[n * NCLS_PAD + tid];
    // ||kappa*mu + f/T||^2 = kappa^2 + 2*kappa*(f.mu)/T + ||f||^2/T^2
    float x2 = k * k + 2.0f * k * d * INV_T + fn2 * INV_T * INV_T;
    float x = sqrtf(fmaxf(x2, 0.0f));
    float l0 = log_ive_nu(x, lt);
    l = l0 + x - (float)NU * logf(x + 1e-20f) - logc[tid];
  }
  lg[tid] = l;
  red[tid] = l;
  __syncthreads();
  for (int st = 64; st > 0; st >>= 1) {
    if (tid < st) red[tid] = fmaxf(red[tid], red[tid + st]);
    __syncthreads();
  }
  float mx = red[0];
  __syncthreads();
  red[tid] = (tid < NCLS) ? __expf(l - mx) : 0.0f;
  __syncthreads();
  for (int st = 64; st > 0; st >>= 1) {
    if (tid < st) red[tid] += red[tid + st];
    __syncthreads();
  }
  if (tid == 0) {
    int lab = labels[n];
    loss[n] = -(lg[lab] - mx - logf(red[0]));
  }
}

// ---------- kernel 5: deterministic mean reduction -----------------------
__global__ __launch_bounds__(256) void reduce_kernel(
    const float* __restrict__ loss, float* __restrict__ out, int N) {
  __shared__ float red[256];
  float s = 0.0f;
  for (int n = threadIdx.x; n < N; n += 256) s += loss[n];
  red[threadIdx.x] = s;
  __syncthreads();
  for (int st = 128; st > 0; st >>= 1) {
    if (threadIdx.x < st) red[threadIdx.x] += red[threadIdx.x + st];
    __syncthreads();
  }
  if (threadIdx.x == 0) out[0] = red[0] / (float)N;
}

// ---------- launch -------------------------------------------------------
extern "C" void kernel_launch(void* const* d_in, const int* in_sizes, int n_in,
                              void* d_out, int out_size, void* d_ws, size_t ws_size,
                              hipStream_t stream) {
  const float* F      = (const float*)d_in[0];   // (N, 256)
  const float* Ave    = (const float*)d_in[1];   // (100, 256)
  const float* Amount = (const float*)d_in[2];   // (100,)
  const int*   labels = (const int*)d_in[3];     // (N, 1)
  float* out = (float*)d_out;
  float* ws  = (float*)d_ws;

  int N = in_sizes[0] / FDIM;                    // 4096

  // workspace layout (floats)
  float* kappa = ws;                             // 112
  float* logc  = ws + NCLS_PAD;                  // 112
  float* mu    = ws + 2 * NCLS_PAD;              // 112*256
  float* dotb  = mu + NCLS_PAD * FDIM;           // N*112
  float* loss  = dotb + (size_t)N * NCLS_PAD;    // N

  class_stats_kernel<<<NCLS_PAD, 256, 0, stream>>>(F, Ave, Amount, labels,
                                                   mu, kappa, N);
  logc_kernel<<<1, 128, 0, stream>>>(kappa, logc);
  dim3 gg(N / 16, NCLS_PAD / 16);
  gemm_dot_kernel<<<gg, 32, 0, stream>>>(F, mu, dotb);
  sample_kernel<<<N, 128, 0, stream>>>(F, dotb, kappa, logc, labels, loss);
  reduce_kernel<<<1, 256, 0, stream>>>(loss, out, N);
}